// DeepHodgeLayer_74036646249034
// MI455X (gfx1250) — compile-verified
//
#include <hip/hip_runtime.h>
#include <math.h>

#define KN 16
#define DM 256
#define SS 3
#define EN 120
#define TN 560
#define TAU 0.0001f

// Flip to 0 if the assembler rejects the gfx1250 async mnemonics.
#define USE_ASYNC_LDS 1

typedef __attribute__((ext_vector_type(2))) float v2f;
typedef __attribute__((ext_vector_type(4))) float v4f;
typedef __attribute__((ext_vector_type(8))) float v8f;

__device__ __forceinline__ v8f wmma4(v2f a, v2f b, v8f c) {
  // V_WMMA_F32_16X16X4_F32 : D = A(16x4) * B(4x16) + C(16x16)
  return __builtin_amdgcn_wmma_f32_16x16x4_f32(false, a, false, b, (short)0, c,
                                               false, false);
}

// ---- async global->LDS copy (CDNA5): 16B per lane, tracked by ASYNCcnt ----
__device__ __forceinline__ void copy_b128_g2l(void* lds, const void* g) {
#if USE_ASYNC_LDS
  // ISA 10.2: generic shared-pointer low 32 bits == wave-relative LDS address.
  unsigned int l = (unsigned int)(unsigned long long)lds;
  asm volatile("global_load_async_to_lds_b128 %0, %1, off"
               :: "v"(l), "v"(g) : "memory");
#else
  *(v4f*)lds = *(const v4f*)g;
#endif
}
__device__ __forceinline__ void wait_async_done() {
#if USE_ASYNC_LDS
  asm volatile("s_wait_asynccnt 0x0" ::: "memory");
#endif
}

// ---------------- compile-time combinatorial tables (K=16) ----------------
struct HodgeTables {
  int ei[EN], ej[EN];
  int edge[KN][KN];
  int tij[TN], tjk[TN], tik[TN];
  int tri[KN][KN][KN]; // sorted triple -> triangle index
};
constexpr HodgeTables make_tables() {
  HodgeTables t{};
  int e = 0;
  for (int i = 0; i < KN; ++i)
    for (int j = i + 1; j < KN; ++j) {
      t.ei[e] = i; t.ej[e] = j; t.edge[i][j] = e; t.edge[j][i] = e; ++e;
    }
  int tt = 0;
  for (int a = 0; a < KN; ++a)
    for (int b = a + 1; b < KN; ++b)
      for (int c = b + 1; c < KN; ++c) {
        t.tij[tt] = t.edge[a][b];
        t.tjk[tt] = t.edge[b][c];
        t.tik[tt] = t.edge[a][c];
        t.tri[a][b][c] = tt; ++tt;
      }
  return t;
}
__constant__ HodgeTables TAB = make_tables();

// ---------------- generic fp32 WMMA GEMM: C = A*B (+bias)(+gelu)(+res) ----
// Block: 128 threads (4 waves). Tile: M=32, N=128 (wave: 2 M-tiles x 2 N-tiles)
// EPI: 0=none, 2=bias+exact GELU, 3=bias+residual
template <int EPI>
__global__ __launch_bounds__(128) void gemm16_kernel(
    const float* __restrict__ A, int lda, const float* __restrict__ Bw, int ldb,
    const float* __restrict__ bias, const float* __restrict__ R, int ldr,
    float* __restrict__ C, int ldc, int M, int N, int Kd) {
  __shared__ float As[32][68]; // stride 68 -> rows 4 banks apart, 16B aligned
  const int tid = threadIdx.x;
  const int lane = tid & 31, w = tid >> 5;
  const int l15 = lane & 15, lh = lane >> 4;
  const int m0 = blockIdx.y * 32;
  const int n0 = blockIdx.x * 128 + w * 32;
  v8f acc00 = {}, acc01 = {}, acc10 = {}, acc11 = {};
  for (int k0 = 0; k0 < Kd; k0 += 64) {
    __syncthreads();
    // stage A chunk 32x64 = 512 float4, 4 per thread (async DMA to LDS)
#pragma unroll
    for (int i = 0; i < 4; ++i) {
      int idx = tid + (i << 7);       // 0..511
      int r = idx >> 4, c4 = idx & 15;
      copy_b128_g2l(&As[r][c4 << 2],
                    &A[(long)(m0 + r) * lda + k0 + (c4 << 2)]);
    }
    wait_async_done();
    __syncthreads();
#pragma unroll
    for (int ks = 0; ks < 16; ++ks) {
      int kb = (ks << 2) + (lh << 1);
      v2f a0, a1;
      a0.x = As[l15][kb];      a0.y = As[l15][kb + 1];
      a1.x = As[16 + l15][kb]; a1.y = As[16 + l15][kb + 1];
      long brow = (long)(k0 + kb) * ldb;
      v2f b0, b1;
      b0.x = Bw[brow + n0 + l15];       b0.y = Bw[brow + ldb + n0 + l15];
      b1.x = Bw[brow + n0 + 16 + l15];  b1.y = Bw[brow + ldb + n0 + 16 + l15];
      acc00 = wmma4(a0, b0, acc00);
      acc01 = wmma4(a0, b1, acc01);
      acc10 = wmma4(a1, b0, acc10);
      acc11 = wmma4(a1, b1, acc11);
    }
  }
#pragma unroll
  for (int mt = 0; mt < 2; ++mt) {
#pragma unroll
    for (int t = 0; t < 2; ++t) {
      v8f acc = mt ? (t ? acc11 : acc10) : (t ? acc01 : acc00);
      int n = n0 + t * 16 + l15;
      float bs = 0.f;
      if (EPI >= 2) bs = bias[n];
#pragma unroll
      for (int r = 0; r < 8; ++r) {
        int m = m0 + mt * 16 + r + (lh << 3);
        float v = acc[r] + bs;
        if (EPI == 2) v = 0.5f * v * (1.0f + erff(v * 0.70710678118654752f));
        if (EPI == 3) v += R[(long)m * ldr + n];
        C[(long)m * ldc + n] = v;
      }
    }
  }
}

// ---------------- LayerNorm (row = 256), vectorized b128 I/O --------------
__global__ __launch_bounds__(256) void ln_kernel(
    const float* __restrict__ X, const float* __restrict__ g,
    const float* __restrict__ b, float* __restrict__ Y, int nrows) {
  int row = blockIdx.x * 8 + (threadIdx.x >> 5);
  if (row >= nrows) return;
  int lane = threadIdx.x & 31;
  const v4f* x4 = (const v4f*)(X + (long)row * DM);
  v4f v[2];
  float s = 0.f;
#pragma unroll
  for (int i = 0; i < 2; ++i) {
    v[i] = x4[lane + 32 * i];
#pragma unroll
    for (int j = 0; j < 4; ++j) s += v[i][j];
  }
#pragma unroll
  for (int o = 16; o; o >>= 1) s += __shfl_xor(s, o, 32);
  float mean = s * (1.0f / DM);
  float vs = 0.f;
#pragma unroll
  for (int i = 0; i < 2; ++i)
#pragma unroll
    for (int j = 0; j < 4; ++j) {
      float d = v[i][j] - mean;
      vs += d * d;
    }
#pragma unroll
  for (int o = 16; o; o >>= 1) vs += __shfl_xor(vs, o, 32);
  float inv = rsqrtf(vs * (1.0f / DM) + 1e-5f);
  const v4f* g4 = (const v4f*)g;
  const v4f* b4 = (const v4f*)b;
  v4f* y4 = (v4f*)(Y + (long)row * DM);
#pragma unroll
  for (int i = 0; i < 2; ++i) {
    int q = lane + 32 * i;
    v4f gg = g4[q], bb = b4[q], o4;
#pragma unroll
    for (int j = 0; j < 4; ++j) o4[j] = (v[i][j] - mean) * inv * gg[j] + bb[j];
    y4[q] = o4;
  }
}

// ---------------- gate: P = xn @ Wg + bg (16 cols) ----------------
__global__ __launch_bounds__(256) void gate_kernel(
    const float* __restrict__ XN, const float* __restrict__ Wg,
    const float* __restrict__ bg, float* __restrict__ P, int nrows) {
  int row = blockIdx.x * 8 + (threadIdx.x >> 5);
  if (row >= nrows) return;
  int lane = threadIdx.x & 31;
  int c = lane & 15, h = lane >> 4;
  const float* x = XN + (long)row * DM;
  float acc = 0.f;
  for (int t = 0; t < 128; ++t) {
    int d = 2 * t + h;
    acc += x[d] * Wg[d * 16 + c];
  }
  acc += __shfl_xor(acc, 16, 32);
  if (lane < 16) P[(long)row * 16 + lane] = acc + bg[lane];
}

// ---------------- affinities: W1[b,s,e], W2[b,s,t] ----------------
__global__ __launch_bounds__(256) void aff_kernel(
    const float* __restrict__ P, const float* __restrict__ mask,
    const float* __restrict__ ls, float* __restrict__ W1,
    float* __restrict__ W2) {
  __shared__ float Ps[KN][17], p2s[KN], ms[KN], Dsq[KN][17], W1s[SS][EN], den[SS];
  int b = blockIdx.x, tid = threadIdx.x;
  Ps[tid >> 4][tid & 15] = P[(long)b * 256 + tid];
  if (tid < KN) ms[tid] = mask[b * KN + tid];
  if (tid < SS) { float sg = expf(2.f * ls[tid]); den[tid] = 2.f * sg + 1e-8f; }
  __syncthreads();
  if (tid < KN) {
    float s = 0.f;
    for (int c = 0; c < 16; ++c) s += Ps[tid][c] * Ps[tid][c];
    p2s[tid] = s;
  }
  __syncthreads();
  {
    int k = tid >> 4, l = tid & 15;
    float dot = 0.f;
    for (int c = 0; c < 16; ++c) dot += Ps[k][c] * Ps[l][c];
    float d = fmaxf(p2s[k] + p2s[l] - 2.f * dot, 0.f) * ms[k] * ms[l];
    Dsq[k][l] = d;
  }
  __syncthreads();
  for (int idx = tid; idx < SS * EN; idx += 256) {
    int s = idx / EN, e = idx - s * EN;
    int i = TAB.ei[e], j = TAB.ej[e];
    float w = expf(-Dsq[i][j] / den[s]) * ms[i] * ms[j];
    W1s[s][e] = w;
    W1[((long)b * SS + s) * EN + e] = w;
  }
  __syncthreads();
  for (int idx = tid; idx < SS * TN; idx += 256) {
    int s = idx / TN, t = idx - s * TN;
    W2[((long)b * SS + s) * TN + t] =
        W1s[s][TAB.tij[t]] * W1s[s][TAB.tjk[t]] * W1s[s][TAB.tik[t]];
  }
}

// ---- hodge core per (b,s): L1 -> AL -> {L0(=Hn), N2(=He), r1(=Re)} ----
__global__ __launch_bounds__(256) void hodge_core_kernel(
    const float* __restrict__ W1g, const float* __restrict__ W2g,
    const float* __restrict__ mask, float* __restrict__ Hn,
    float* __restrict__ He, float* __restrict__ Re) {
  extern __shared__ float sm[];
  float* L1 = sm;                 // 120*121
  float* AL = L1 + 120 * 121;     // 16*121
  float* W1s = AL + 16 * 121;     // 120
  float* W2s = W1s + 120;         // 560
  float* ms = W2s + 560;          // 16
  int bs = blockIdx.x, b = bs / 3, tid = threadIdx.x;
  for (int i = tid; i < EN; i += 256) W1s[i] = W1g[(long)bs * EN + i];
  for (int i = tid; i < TN; i += 256) W2s[i] = W2g[(long)bs * TN + i];
  if (tid < KN) ms[tid] = mask[b * KN + tid];
  __syncthreads();
  for (int idx = tid; idx < EN * EN; idx += 256) {
    int e = idx / EN, f = idx - e * EN;
    int i = TAB.ei[e], j = TAB.ej[e];
    float val;
    if (e == f) {
      float su = 0.f;
      for (int r = 0; r < KN; ++r) {
        if (r == i || r == j) continue;
        int a0, a1, a2;
        if (r < i)      { a0 = r; a1 = i; a2 = j; }
        else if (r < j) { a0 = i; a1 = r; a2 = j; }
        else            { a0 = i; a1 = j; a2 = r; }
        su += W2s[TAB.tri[a0][a1][a2]];
      }
      val = TAU + ms[i] + ms[j] + su;
    } else {
      int p = TAB.ei[f], q = TAB.ej[f];
      int v = (i == p || i == q) ? i : ((j == p || j == q) ? j : -1);
      if (v < 0) val = 0.f;
      else {
        float s1 = (v == i) ? -1.f : 1.f;
        float s2 = (v == p) ? -1.f : 1.f;
        float td = s1 * s2 * ms[v];
        int wv = (v == p) ? q : p;
        int a0, a1, a2;
        if (wv < i)      { a0 = wv; a1 = i;  a2 = j; }
        else if (wv < j) { a0 = i;  a1 = wv; a2 = j; }
        else             { a0 = i;  a1 = j;  a2 = wv; }
        int t = TAB.tri[a0][a1][a2];
        float se = (i == a0 && j == a2) ? -1.f : 1.f;
        float sf = (p == a0 && q == a2) ? -1.f : 1.f;
        val = td + se * sf * W2s[t];
      }
    }
    L1[e * 121 + f] = val;
  }
  __syncthreads();
  for (int idx = tid; idx < KN * EN; idx += 256) {  // AL = ABS @ L1
    int k = idx / EN, f = idx - k * EN;
    float s = 0.f;
    for (int r = 0; r < KN; ++r)
      if (r != k) s += L1[TAB.edge[k][r] * 121 + f];
    AL[k * 121 + f] = s;
  }
  __syncthreads();
  {
    int k = tid >> 4, v = tid & 15;
    float s = 0.f;                                  // N2 = AL @ ABS^T
    for (int r = 0; r < KN; ++r)
      if (r != v) s += AL[k * 121 + TAB.edge[v][r]];
    He[(long)bs * 256 + k * 16 + v] = s;
    float l0;
    if (k == v) {
      l0 = TAU;
      for (int r = 0; r < KN; ++r) if (r != k) l0 += W1s[TAB.edge[k][r]];
    } else l0 = -W1s[TAB.edge[k][v]];
    Hn[(long)bs * 256 + k * 16 + v] = l0;
  }
  if (tid < KN) {
    float s = 0.f;
    for (int f = 0; f < EN; ++f) s += AL[tid * 121 + f];
    Re[(long)bs * KN + tid] = s;
  }
}

// ---- apply: xo = x + sum_s (Hn@xn)@Wvo_s + (He@xn)@Weo_s + biases ----
__global__ __launch_bounds__(256) void apply_hodge_kernel(
    const float* __restrict__ x, const float* __restrict__ XN,
    const float* __restrict__ Hn, const float* __restrict__ He,
    const float* __restrict__ Re, const float* __restrict__ Wvo,
    const float* __restrict__ Weo, const float* __restrict__ beo,
    const float* __restrict__ bsum, float* __restrict__ xo) {
  __shared__ float XNs[16][260];
  __shared__ float TT[16][260];
  __shared__ float HH[16][17];
  int b = blockIdx.x, tid = threadIdx.x;
  int lane = tid & 31, w = tid >> 5;
  int l15 = lane & 15, lh = lane >> 4;
  // stage xn[b] (16x256) via async b128: 1024 float4, 4 per thread
#pragma unroll
  for (int i = 0; i < 4; ++i) {
    int f4 = tid + (i << 8);
    int r = f4 >> 6, c4 = f4 & 63;
    copy_b128_g2l(&XNs[r][c4 << 2], &XN[(long)b * 4096 + (f4 << 2)]);
  }
  wait_async_done();
  int n0 = w * 32;
  v8f acc0 = {}; v8f acc1 = {};
  for (int s = 0; s < SS; ++s) {
    for (int which = 0; which < 2; ++which) {
      const float* H = (which == 0 ? Hn : He) + ((long)b * 3 + s) * 256;
      const float* Wm = (which == 0 ? Wvo : Weo) + (long)s * 65536;
      __syncthreads();
      HH[tid >> 4][tid & 15] = H[tid];
      __syncthreads();
      v8f c0 = {}; v8f c1 = {};
#pragma unroll
      for (int ks = 0; ks < 4; ++ks) {   // T = H @ xn (K=16)
        int kb = (ks << 2) + (lh << 1);
        v2f a; a.x = HH[l15][kb]; a.y = HH[l15][kb + 1];
        v2f b0, b1;
        b0.x = XNs[kb][n0 + l15];       b0.y = XNs[kb + 1][n0 + l15];
        b1.x = XNs[kb][n0 + 16 + l15];  b1.y = XNs[kb + 1][n0 + 16 + l15];
        c0 = wmma4(a, b0, c0);
        c1 = wmma4(a, b1, c1);
      }
#pragma unroll
      for (int r = 0; r < 8; ++r) {
        TT[r + (lh << 3)][n0 + l15] = c0[r];
        TT[r + (lh << 3)][n0 + 16 + l15] = c1[r];
      }
      __syncthreads();
#pragma unroll 4
      for (int ks = 0; ks < 64; ++ks) {  // acc += T @ Wm (K=256)
        int kb = (ks << 2) + (lh << 1);
        v2f a; a.x = TT[l15][kb]; a.y = TT[l15][kb + 1];
        v2f b0, b1;
        b0.x = Wm[kb * 256 + n0 + l15];
        b0.y = Wm[(kb + 1) * 256 + n0 + l15];
        b1.x = Wm[kb * 256 + n0 + 16 + l15];
        b1.y = Wm[(kb + 1) * 256 + n0 + 16 + l15];
        acc0 = wmma4(a, b0, acc0);
        acc1 = wmma4(a, b1, acc1);
      }
    }
  }
#pragma unroll
  for (int t = 0; t < 2; ++t) {
    v8f acc = t ? acc1 : acc0;
    int n = n0 + t * 16 + l15;
    float bb = bsum[n];
    float e0 = beo[n], e1 = beo[256 + n], e2 = beo[512 + n];
#pragma unroll
    for (int r = 0; r < 8; ++r) {
      int m = r + (lh << 3);
      long rowoff = (long)b * 16 + m;
      float v = acc[r] + bb + Re[((long)b * 3 + 0) * 16 + m] * e0 +
                Re[((long)b * 3 + 1) * 16 + m] * e1 +
                Re[((long)b * 3 + 2) * 16 + m] * e2 + x[rowoff * 256 + n];
      xo[rowoff * 256 + n] = v;
    }
  }
}

// ---------------- fold biases through Wo ----------------
__global__ __launch_bounds__(256) void bias_prep_kernel(
    const float* __restrict__ bv, const float* __restrict__ be,
    const float* __restrict__ Wo, const float* __restrict__ bo,
    float* __restrict__ beo, float* __restrict__ bsum) {
  int c = threadIdx.x;
  float bs = bo[c];
  for (int s = 0; s < SS; ++s) {
    float av = 0.f, ae = 0.f;
    for (int m = 0; m < 256; ++m) {
      av += bv[s * 256 + m] * Wo[(s * 256 + m) * 256 + c];
      ae += be[s * 256 + m] * Wo[(768 + s * 256 + m) * 256 + c];
    }
    beo[s * 256 + c] = ae;
    bs += TAU * av;  // node-bias term: L0 row-sums are exactly tau
  }
  bsum[c] = bs;
}

extern "C" void kernel_launch(void* const* d_in, const int* in_sizes, int n_in,
                              void* d_out, int out_size, void* d_ws,
                              size_t ws_size, hipStream_t stream) {
  (void)n_in; (void)out_size; (void)ws_size;
  const float* x    = (const float*)d_in[0];
  const float* mask = (const float*)d_in[1];
  const float* Wg   = (const float*)d_in[2];
  const float* bg   = (const float*)d_in[3];
  const float* ls   = (const float*)d_in[4];
  const float* Wv   = (const float*)d_in[5];
  const float* bv   = (const float*)d_in[6];
  const float* We   = (const float*)d_in[7];
  const float* be   = (const float*)d_in[8];
  const float* Wo   = (const float*)d_in[9];
  const float* bo   = (const float*)d_in[10];
  const float* g1   = (const float*)d_in[11];
  const float* b1   = (const float*)d_in[12];
  const float* g2   = (const float*)d_in[13];
  const float* b2   = (const float*)d_in[14];
  const float* Wf1  = (const float*)d_in[15];
  const float* bf1  = (const float*)d_in[16];
  const float* Wf2  = (const float*)d_in[17];
  const float* bf2  = (const float*)d_in[18];
  float* out = (float*)d_out;

  const long B = in_sizes[0] / (KN * DM);   // 1024
  const long rows = B * KN;                 // 16384

  float* w = (float*)d_ws;
  size_t o = 0;
  auto alloc = [&](size_t n) { float* p = w + o; o += n; return p; };
  float* xn   = alloc(rows * DM);
  float* P    = alloc(rows * KN);
  float* W1   = alloc(B * SS * EN);
  float* W2   = alloc(B * SS * TN);
  float* Hn   = alloc(B * SS * 256);
  float* He   = alloc(B * SS * 256);
  float* Re   = alloc(B * SS * KN);
  float* Wvo  = alloc(SS * 65536);
  float* Weo  = alloc(SS * 65536);
  float* beo  = alloc(SS * 256);
  float* bsum = alloc(256);
  float* xo   = alloc(rows * DM);
  float* h    = alloc(rows * DM);
  float* f    = alloc(rows * 1024);

  // 1) fold weights: Wvo_s = Wv_s @ Wo[s*256:+256], Weo_s = We_s @ Wo[768+s*256:+256]
  for (int s = 0; s < SS; ++s) {
    gemm16_kernel<0><<<dim3(2, 8), 128, 0, stream>>>(
        Wv + s * 256, 768, Wo + (size_t)s * 256 * 256, 256, nullptr, nullptr, 0,
        Wvo + (size_t)s * 65536, 256, 256, 256, 256);
    gemm16_kernel<0><<<dim3(2, 8), 128, 0, stream>>>(
        We + s * 256, 768, Wo + (size_t)(768 + s * 256) * 256, 256, nullptr,
        nullptr, 0, Weo + (size_t)s * 65536, 256, 256, 256, 256);
  }
  bias_prep_kernel<<<1, 256, 0, stream>>>(bv, be, Wo, bo, beo, bsum);

  // 2) LN1, gate, affinities
  ln_kernel<<<dim3((rows + 7) / 8), 256, 0, stream>>>(x, g1, b1, xn, (int)rows);
  gate_kernel<<<dim3((rows + 7) / 8), 256, 0, stream>>>(xn, Wg, bg, P, (int)rows);
  aff_kernel<<<dim3(B), 256, 0, stream>>>(P, mask, ls, W1, W2);

  // 3) hodge core: L1 in dynamic LDS (68.6 KB)
  size_t lds = (size_t)(120 * 121 + 16 * 121 + 120 + 560 + 16) * sizeof(float);
  hodge_core_kernel<<<dim3(B * SS), 256, lds, stream>>>(W1, W2, mask, Hn, He, Re);

  // 4) apply folded 16x16 operators (all WMMA)
  apply_hodge_kernel<<<dim3(B), 256, 0, stream>>>(x, xn, Hn, He, Re, Wvo, Weo,
                                                  beo, bsum, xo);

  // 5) FFN: h = LN2(xo); f = gelu(h@Wf1+bf1); out = xo + f@Wf2 + bf2
  ln_kernel<<<dim3((rows + 7) / 8), 256, 0, stream>>>(xo, g2, b2, h, (int)rows);
  gemm16_kernel<2><<<dim3(8, rows / 32), 128, 0, stream>>>(
      h, 256, Wf1, 1024, bf1, nullptr, 0, f, 1024, (int)rows, 1024, 256);
  gemm16_kernel<3><<<dim3(2, rows / 32), 128, 0, stream>>>(
      f, 1024, Wf2, 256, bf2, xo, 256, out, 256, (int)rows, 256, 1024);
}